// CossimBce_23270132810103
// MI455X (gfx1250) — compile-verified
//
#include <hip/hip_runtime.h>
#include <hip/hip_bf16.h>
#include <math.h>

// CDNA5 wave32 WMMA fp32: D(16x16) = A(16x4) * B(4x16) + C
typedef __attribute__((ext_vector_type(2))) float v2f;
typedef __attribute__((ext_vector_type(8))) float v8f;

#define BATCH 8
#define NDIM  2048
#define MDIM  2048
#define CDIM  128
#define EPSV  1e-8f

// ---------------- norms: n1[b,n] = ||x1[b,n,:]||, n2[b,m] = ||x2[b,m,:]|| -------------
__global__ void __launch_bounds__(256)
cossim_norms_kernel(const float* __restrict__ x1, const float* __restrict__ x2,
                    float* __restrict__ n1, float* __restrict__ n2) {
    const int lane = threadIdx.x & 31;
    const int wave = threadIdx.x >> 5;
    const int row  = blockIdx.x * 8 + wave;
    const int R1 = BATCH * NDIM;
    const int R2 = BATCH * MDIM;
    const float* src;
    float* dst;
    if (row < R1)            { src = x1 + (size_t)row * CDIM;        dst = n1 + row; }
    else if (row < R1 + R2)  { int r = row - R1; src = x2 + (size_t)r * CDIM; dst = n2 + r; }
    else return;
    // 32 lanes x float4 = 128 = C
    float4 v = ((const float4*)src)[lane];
    float s = v.x * v.x + v.y * v.y + v.z * v.z + v.w * v.w;
#pragma unroll
    for (int off = 16; off >= 1; off >>= 1)
        s += __shfl_xor(s, off, 32);
    if (lane == 0) *dst = sqrtf(s);
}

// ---------------- main: fp32 WMMA GEMM + fused cos/mask/nll epilogue -------------
__global__ void __launch_bounds__(256)
cossim_bce_main_kernel(const int*   __restrict__ z,
                       const float* __restrict__ x1,
                       const float* __restrict__ x2,
                       const float* __restrict__ tptr,
                       const float* __restrict__ bptr,
                       const float* __restrict__ n1,
                       const float* __restrict__ n2,
                       float* __restrict__ cos_out,
                       float* __restrict__ mask_out,
                       float* __restrict__ part_sum,
                       unsigned* __restrict__ part_cnt) {
    // block tile: 64 (n) x 128 (m); wave tile: 32x32 = 2x2 WMMA tiles of 16x16
    const int MB = MDIM / 128;            // 16
    const int NB = NDIM / 64;             // 32
    int bidx = blockIdx.x;
    const int bm = bidx % MB;  bidx /= MB;
    const int bn = bidx % NB;
    const int bb = bidx / NB;

    const int lane = threadIdx.x & 31;
    const int wave = threadIdx.x >> 5;
    const int wn = wave & 1;              // 2 waves along n
    const int wm = wave >> 1;             // 4 waves along m

    const int n0 = bn * 64  + wn * 32;
    const int m0 = bm * 128 + wm * 32;

    const float* x1b = x1 + (size_t)bb * NDIM * CDIM;
    const float* x2b = x2 + (size_t)bb * MDIM * CDIM;

    // f32 16x4 A / 4x16 B fragment layout: lane holds row (lane&15),
    // k-pair offset 2*(lane>>4); consecutive VGPRs = consecutive k.
    const int lrow = lane & 15;
    const int koff = (lane >> 4) << 1;

    const float* a0p = x1b + (size_t)(n0      + lrow) * CDIM + koff;
    const float* a1p = x1b + (size_t)(n0 + 16 + lrow) * CDIM + koff;
    const float* b0p = x2b + (size_t)(m0      + lrow) * CDIM + koff;
    const float* b1p = x2b + (size_t)(m0 + 16 + lrow) * CDIM + koff;

    v8f acc[2][2] = {};

#pragma unroll 4
    for (int k = 0; k < CDIM; k += 4) {
        v2f a0 = *(const v2f*)(a0p + k);
        v2f a1 = *(const v2f*)(a1p + k);
        v2f b0 = *(const v2f*)(b0p + k);
        v2f b1 = *(const v2f*)(b1p + k);
        acc[0][0] = __builtin_amdgcn_wmma_f32_16x16x4_f32(false, a0, false, b0, (short)0, acc[0][0], false, false);
        acc[0][1] = __builtin_amdgcn_wmma_f32_16x16x4_f32(false, a0, false, b1, (short)0, acc[0][1], false, false);
        acc[1][0] = __builtin_amdgcn_wmma_f32_16x16x4_f32(false, a1, false, b0, (short)0, acc[1][0], false, false);
        acc[1][1] = __builtin_amdgcn_wmma_f32_16x16x4_f32(false, a1, false, b1, (short)0, acc[1][1], false, false);
    }

    // ------- fused epilogue (HBM-bound part: z read, cos/mask write) -------
    const float tval = tptr[0];
    const float bval = bptr[0];
    const float* n1b = n1 + bb * NDIM;
    const float* n2b = n2 + bb * MDIM;
    const size_t plane = (size_t)bb * NDIM * MDIM;

    float    lsum = 0.0f;
    unsigned lcnt = 0u;
    const int rowsel = (lane >> 4) << 3;  // C/D layout: M = vgpr + 8*(lane>=16)

#pragma unroll
    for (int ti = 0; ti < 2; ++ti) {
#pragma unroll
        for (int tj = 0; tj < 2; ++tj) {
            const int nt = n0 + 16 * ti;
            const int mt = m0 + 16 * tj;
            const int m  = mt + lrow;     // C/D layout: N = lane & 15
            const float nn2 = n2b[m];
#pragma unroll
            for (int v = 0; v < 8; ++v) {
                const int n = nt + v + rowsel;
                const float d     = acc[ti][tj][v];
                const float denom = fmaxf(n1b[n] * nn2, EPSV);
                const float cs    = d / denom;
                const size_t idx  = plane + (size_t)n * MDIM + m;
                cos_out[idx] = cs;
                const int zv = z[idx];
                mask_out[idx] = (zv != 0) ? 1.0f : 0.0f;
                if (zv != 0) {
                    const float y  = (float)zv * (tval * cs - bval);
                    const float xn = -y;                        // nll = softplus(-y)
                    const float nll = fmaxf(xn, 0.0f) + log1pf(expf(-fabsf(xn)));
                    lsum += nll;
                    ++lcnt;
                }
            }
        }
    }

    // deterministic per-block reduction
    __shared__ float    ssum[256];
    __shared__ unsigned scnt[256];
    ssum[threadIdx.x] = lsum;
    scnt[threadIdx.x] = lcnt;
    __syncthreads();
#pragma unroll
    for (int s = 128; s > 0; s >>= 1) {
        if ((int)threadIdx.x < s) {
            ssum[threadIdx.x] += ssum[threadIdx.x + s];
            scnt[threadIdx.x] += scnt[threadIdx.x + s];
        }
        __syncthreads();
    }
    if (threadIdx.x == 0) {
        part_sum[blockIdx.x] = ssum[0];
        part_cnt[blockIdx.x] = scnt[0];
    }
}

// ---------------- finalize: fixed-order reduction -> loss -------------
__global__ void __launch_bounds__(256)
cossim_finalize_kernel(const float* __restrict__ part_sum,
                       const unsigned* __restrict__ part_cnt,
                       int nparts, float* __restrict__ loss) {
    __shared__ float    ssum[256];
    __shared__ unsigned scnt[256];
    float s = 0.0f;
    unsigned c = 0u;
    for (int i = threadIdx.x; i < nparts; i += 256) {
        s += part_sum[i];
        c += part_cnt[i];
    }
    ssum[threadIdx.x] = s;
    scnt[threadIdx.x] = c;
    __syncthreads();
#pragma unroll
    for (int st = 128; st > 0; st >>= 1) {
        if ((int)threadIdx.x < st) {
            ssum[threadIdx.x] += ssum[threadIdx.x + st];
            scnt[threadIdx.x] += scnt[threadIdx.x + st];
        }
        __syncthreads();
    }
    if (threadIdx.x == 0)
        loss[0] = ssum[0] / (float)scnt[0];
}

extern "C" void kernel_launch(void* const* d_in, const int* in_sizes, int n_in,
                              void* d_out, int out_size, void* d_ws, size_t ws_size,
                              hipStream_t stream) {
    const int*   z  = (const int*)d_in[0];
    const float* x1 = (const float*)d_in[1];
    const float* x2 = (const float*)d_in[2];
    const float* t  = (const float*)d_in[3];
    const float* b  = (const float*)d_in[4];

    float* out      = (float*)d_out;
    float* loss     = out;                                    // [1]
    float* cos_out  = out + 1;                                // [B,N,M]
    float* mask_out = out + 1 + (size_t)BATCH * NDIM * MDIM;  // [B,N,M]

    // workspace layout
    float* n1       = (float*)d_ws;                 // B*N
    float* n2       = n1 + BATCH * NDIM;            // B*M
    const int nblocks = BATCH * (NDIM / 64) * (MDIM / 128);   // 4096
    float*    part_sum = n2 + BATCH * MDIM;         // nblocks
    unsigned* part_cnt = (unsigned*)(part_sum + nblocks);     // nblocks

    const int rows = BATCH * NDIM + BATCH * MDIM;
    cossim_norms_kernel<<<(rows + 7) / 8, 256, 0, stream>>>(x1, x2, n1, n2);
    cossim_bce_main_kernel<<<nblocks, 256, 0, stream>>>(z, x1, x2, t, b, n1, n2,
                                                        cos_out, mask_out,
                                                        part_sum, part_cnt);
    cossim_finalize_kernel<<<1, 256, 0, stream>>>(part_sum, part_cnt, nblocks, loss);
}